// KroneckerProduct_19499151524244
// MI455X (gfx1250) — compile-verified
//
#include <hip/hip_runtime.h>
#include <hip/hip_bf16.h>

// Batched Kronecker product on gfx1250 (MI455X):
//   out[b, i*64+p, j*64+q] = A[b,i,j] * B[b,p,q]
//   A,B: [8,64,64] f32 ; out: [8,4096,4096] f32  (537 MB of stores -> BW bound)
//
// Tile: fixed (b, j, p): D[M,N] = A[b, i0+M, j] * B[b, p, q0*16+N] is a rank-1
// outer product -> one V_WMMA_F32_16X16X4_F32 per 256 outputs, K=0 only.
//
// Operand trick: A-operand lanes 16-31 of VGPR0 (the K=2 column) are forced to
// zero once, so the B-operand's K=2 row (its lanes 16-31) is multiplied by 0
// and needs NO masking -> B row loads are uniform, non-divergent (addresses use
// lane&15, so hi lanes re-read valid finite floats; no NaN fabrication).

typedef __attribute__((ext_vector_type(2))) float v2f;
typedef __attribute__((ext_vector_type(8))) float v8f;

__global__ __launch_bounds__(256) void kron_wmma_kernel(const float* __restrict__ A,
                                                        const float* __restrict__ B,
                                                        float* __restrict__ out) {
  // blockIdx.x in [0, 8*4*64): decode (b, i0 block, j)
  const int j  = blockIdx.x & 63;          // A column / output super-column
  const int i0 = ((blockIdx.x >> 6) & 3) * 16;
  const int b  = blockIdx.x >> 8;

  const int lane = threadIdx.x & 31;       // wave32
  const int wave = threadIdx.x >> 5;       // 0..7
  const int n    = lane & 15;              // N (and duplicate-read index)
  const int mbase = (lane >> 4) * 8;       // lanes 16-31 hold M=r+8 in D

  const float* __restrict__ Ab = A + b * 64 * 64;
  const float* __restrict__ Bb = B + b * 64 * 64;
  float* __restrict__ Ob = out + (size_t)b * 4096 * 4096;

  // u[M] = A[b, i0+M, j] in lanes 0-15 (K=0). Lanes 16-31 (K=2 column) MUST be
  // zero -- this is the one mask in the kernel; it also zeroes B's K=2 row's
  // contribution, making B loads mask-free.
  const float aval = Ab[(i0 + n) * 64 + j];
  v2f a;
  a.x = (lane < 16) ? aval : 0.0f;
  a.y = 0.0f;                              // K=1 / K=3 slots

  // Each wave: 8 p-values x 4 q0 tiles; u reused across all 32 WMMAs.
  for (int pt = 0; pt < 8; ++pt) {
    const int p = wave * 8 + pt;           // B row / output row offset

    // Clause all 4 q0 row segments up front (uniform, no divergence).
    float bval[4];
    #pragma unroll
    for (int q0 = 0; q0 < 4; ++q0)
      bval[q0] = Bb[p * 64 + q0 * 16 + n];

    #pragma unroll
    for (int q0 = 0; q0 < 4; ++q0) {
      v2f bv;
      bv.x = bval[q0];                     // hi lanes = K=2 row: killed by a.x==0
      bv.y = 0.0f;                         // K=1 / K=3 slots

      v8f c = {};
      // 8 args: (neg_a, A, neg_b, B, c_mod, C, reuse_a, reuse_b)
      c = __builtin_amdgcn_wmma_f32_16x16x4_f32(
          /*neg_a=*/false, a, /*neg_b=*/false, bv,
          /*c_mod=*/(short)0, c, /*reuse_a=*/false, /*reuse_b=*/false);

      // Store D: row I = (i0+M)*64 + p, col J = j*64 + q0*16 + N.
      // 8 b32 stores, each coalescing into two contiguous 64B segments.
      const size_t colBase = (size_t)(j * 64 + q0 * 16 + n);
      #pragma unroll
      for (int r = 0; r < 8; ++r) {
        const int M = mbase + r;
        const size_t row = (size_t)((i0 + M) * 64 + p);
        Ob[row * 4096 + colBase] = c[r];
      }
    }
  }
}

extern "C" void kernel_launch(void* const* d_in, const int* in_sizes, int n_in,
                              void* d_out, int out_size, void* d_ws, size_t ws_size,
                              hipStream_t stream) {
  const float* A = (const float*)d_in[0];   // [8,64,64]
  const float* B = (const float*)d_in[1];   // [8,64,64]
  float* out = (float*)d_out;               // [8,4096,4096]

  dim3 grid(8 * 4 * 64);                    // (b, i0/16, j)
  dim3 block(256);                          // 8 waves of 32
  kron_wmma_kernel<<<grid, block, 0, stream>>>(A, B, out);
}